// TileWarping_64802466562421
// MI455X (gfx1250) — compile-verified
//
#include <hip/hip_runtime.h>

// Reference shapes:
//   tile_plane: (4,3,96,192) f32   [d, dx, dy]
//   fea_l/fea_r: (4,32,384,768) f32
//   out: (4,48,96,192) f32  (3 k-hypotheses x 16 pixel-unshuffle channels)

#define FB 4
#define FC 32
#define FH 384
#define FW 768
#define TH 96
#define TW 192
#define OCH 48
#define NPIX (FB * FH * FW)          // 1,179,648 full-res pixels
#define OUT_KSTRIDE (16 * TH * TW)   // channel-group stride between k's
#define CH_STRIDE (FH * FW)

typedef float v2f __attribute__((ext_vector_type(2)));
typedef float v8f __attribute__((ext_vector_type(8)));

#if defined(__HIP_DEVICE_COMPILE__) && __has_builtin(__builtin_amdgcn_wmma_f32_16x16x4_f32)
#define USE_WMMA 1
#else
#define USE_WMMA 0
#endif

__global__ __launch_bounds__(256) void tile_warp_cv(
    const float* __restrict__ tp,
    const float* __restrict__ fl,
    const float* __restrict__ fr,
    float* __restrict__ out)
{
    const int lane = threadIdx.x & 31;
    const int wave = (blockIdx.x << 3) | (threadIdx.x >> 5);
    const int pixBase = wave << 4;          // 16 pixels per wave, same image row
    const int m = lane & 15;                // pixel slot (matrix row M)
    const int p = pixBase + m;

    const int x  = p % FW;
    const int yz = p / FW;
    const int y  = yz % FH;
    const int b  = yz / FH;

    // ---- disparity plane for k = 0 (k shifts are exact integer tap shifts) ----
    const int ty = y >> 2, tx = x >> 2;
    const int tpb = (b * 3 * TH + ty) * TW + tx;
    const float d0  = tp[tpb];
    const float dxv = tp[tpb + TH * TW];
    const float dyv = tp[tpb + 2 * TH * TW];
    const float acoef = (float)(x & 3) - 1.5f;
    const float bcoef = (float)(y & 3) - 1.5f;
    const float px = (float)x - (d0 + acoef * dxv + bcoef * dyv);
    const float xf = floorf(px);
    const float w1 = px - xf;               // horizontal lerp fraction (shared by all k)
    const float w0 = 1.0f - w1;
    const int   x0 = (int)xf;

    // 4 taps x0-1..x0+2 cover k=+1 (t0,t1), k=0 (t1,t2), k=-1 (t2,t3); zero-pad OOB
    int   xc[4];
    float vm[4];
#pragma unroll
    for (int t = 0; t < 4; ++t) {
        const int xi = x0 - 1 + t;
        vm[t] = (xi >= 0 && xi < FW) ? 1.0f : 0.0f;
        xc[t] = xi < 0 ? 0 : (xi > FW - 1 ? FW - 1 : xi);
    }

    const int rowL = (b * FC * FH + y) * FW;   // + c*CH_STRIDE + x

#if USE_WMMA
    // A fragment (16x4 f32): lanes 0-15 carry K=0,1 ; lanes 16-31 carry K=2,3.
    // So lane L and L+16 handle the same pixel, different channel pairs.
    // B = all-ones 4x16 => D[m][n] = C[m][n] + sum_K A[m][K] (every column equal).
    v8f accN = {0,0,0,0,0,0,0,0};   // k = -1
    v8f accZ = {0,0,0,0,0,0,0,0};   // k =  0
    v8f accP = {0,0,0,0,0,0,0,0};   // k = +1
    const v2f onesB = {1.0f, 1.0f};
    const int chOfs = (lane >> 4) << 1;        // 0 for lanes 0-15, 2 for lanes 16-31

#pragma unroll
    for (int cc = 0; cc < 8; ++cc) {
        const int chA = (cc << 2) + chOfs;
        float aN[2], aZ[2], aP[2];
#pragma unroll
        for (int s = 0; s < 2; ++s) {
            const int rb = rowL + (chA + s) * CH_STRIDE;
            const float flv = fl[rb + x];
            const float t0 = fr[rb + xc[0]] * vm[0];
            const float t1 = fr[rb + xc[1]] * vm[1];
            const float t2 = fr[rb + xc[2]] * vm[2];
            const float t3 = fr[rb + xc[3]] * vm[3];
            aN[s] = fabsf(flv - (w0 * t2 + w1 * t3));   // k=-1: px+1
            aZ[s] = fabsf(flv - (w0 * t1 + w1 * t2));   // k= 0
            aP[s] = fabsf(flv - (w0 * t0 + w1 * t1));   // k=+1: px-1
        }
        v2f AN = {aN[0], aN[1]};
        v2f AZ = {aZ[0], aZ[1]};
        v2f AP = {aP[0], aP[1]};
        accN = __builtin_amdgcn_wmma_f32_16x16x4_f32(false, AN, false, onesB,
                                                     (short)0, accN, false, false);
        accZ = __builtin_amdgcn_wmma_f32_16x16x4_f32(false, AZ, false, onesB,
                                                     (short)0, accZ, false, false);
        accP = __builtin_amdgcn_wmma_f32_16x16x4_f32(false, AP, false, onesB,
                                                     (short)0, accP, false, false);
    }

    // D layout: vgpr i, lanes 0-15 -> pixel i ; lanes 16-31 -> pixel 8+i.
    // All columns identical (B was all ones), so lanes i and 16+i write pixels i / 8+i.
#pragma unroll
    for (int i = 0; i < 8; ++i) {
        if ((lane & 15) == i) {
            const int ps  = pixBase + i + ((lane >> 4) << 3);
            const int xs  = ps % FW;
            const int yzs = ps / FW;
            const int ys  = yzs % FH;
            const int bs  = yzs / FH;
            const int ob  = ((bs * OCH + (ys & 3) * 4 + (xs & 3)) * TH + (ys >> 2)) * TW
                            + (xs >> 2);
            out[ob]                   = accN[i];
            out[ob + OUT_KSTRIDE]     = accZ[i];
            out[ob + 2 * OUT_KSTRIDE] = accP[i];
        }
    }
#else
    // Fallback (no wmma builtin in this pass): per-lane partial over its channel
    // pair, combine halves with a lane^16 shuffle.
    float sN = 0.0f, sZ = 0.0f, sP = 0.0f;
    const int chOfs = (lane >> 4) << 1;
#pragma unroll
    for (int cc = 0; cc < 8; ++cc) {
        const int chA = (cc << 2) + chOfs;
#pragma unroll
        for (int s = 0; s < 2; ++s) {
            const int rb = rowL + (chA + s) * CH_STRIDE;
            const float flv = fl[rb + x];
            const float t0 = fr[rb + xc[0]] * vm[0];
            const float t1 = fr[rb + xc[1]] * vm[1];
            const float t2 = fr[rb + xc[2]] * vm[2];
            const float t3 = fr[rb + xc[3]] * vm[3];
            sN += fabsf(flv - (w0 * t2 + w1 * t3));
            sZ += fabsf(flv - (w0 * t1 + w1 * t2));
            sP += fabsf(flv - (w0 * t0 + w1 * t1));
        }
    }
    sN += __shfl(sN, lane ^ 16);
    sZ += __shfl(sZ, lane ^ 16);
    sP += __shfl(sP, lane ^ 16);
    if (lane < 16) {
        const int ob = ((b * OCH + (y & 3) * 4 + (x & 3)) * TH + ty) * TW + tx;
        out[ob]                   = sN;
        out[ob + OUT_KSTRIDE]     = sZ;
        out[ob + 2 * OUT_KSTRIDE] = sP;
    }
#endif
}

extern "C" void kernel_launch(void* const* d_in, const int* in_sizes, int n_in,
                              void* d_out, int out_size, void* d_ws, size_t ws_size,
                              hipStream_t stream) {
    const float* tp = (const float*)d_in[0];   // tile_plane (4,3,96,192)
    const float* fl = (const float*)d_in[1];   // fea_l (4,32,384,768)
    const float* fr = (const float*)d_in[2];   // fea_r (4,32,384,768)
    float* out = (float*)d_out;                // (4,48,96,192)

    // 16 pixels per wave, 8 waves per block -> NPIX / 128 blocks (exact)
    const int blocks = NPIX / (16 * 8);        // 9216
    tile_warp_cv<<<blocks, 256, 0, stream>>>(tp, fl, fr, out);
}